// TgvPdhgNet_87557203296718
// MI455X (gfx1250) — compile-verified
//
#include <hip/hip_runtime.h>
#include <stdint.h>

// ---------------- problem geometry (fixed by the reference) ----------------
#define Hh     256
#define Ww     256
#define HW     (Hh * Ww)
#define NELEM  (2 * HW)          // B*H*W = 131072
#define TPB    256               // 8 wave32 waves per block
#define NBLK   128               // 128 strips of 4 rows (2 images * 64 strips)
#define ROWS   4                 // rows per block
#define STG    ((ROWS + 1) * Ww) // 1280 staged words per plane (4 rows + 1 halo row)

// ---------------- CDNA5 async global->LDS copy (per-lane, ASYNCcnt) ----------------
typedef __attribute__((address_space(1))) int GI32;  // global int
typedef __attribute__((address_space(3))) int LI32;  // LDS int

__device__ __forceinline__ void async_copy_f32(float* lds_dst, const float* g_src) {
#if __has_builtin(__builtin_amdgcn_global_load_async_to_lds_b32)
    __builtin_amdgcn_global_load_async_to_lds_b32((GI32*)g_src, (LI32*)lds_dst, 0, 0);
#else
    unsigned loff = (unsigned)(uintptr_t)lds_dst;  // low 32 bits of flat LDS addr = LDS offset
    asm volatile("global_load_async_to_lds_b32 %0, %1, off"
                 :: "v"(loff), "v"(g_src) : "memory");
#endif
}

__device__ __forceinline__ void wait_async0() {
#if __has_builtin(__builtin_amdgcn_s_wait_asynccnt)
    __builtin_amdgcn_s_wait_asynccnt(0);
#else
    asm volatile("s_wait_asynccnt 0" ::: "memory");
#endif
}

// ---------------- CDNA5 Tensor Data Mover (bulk 2D tile, TENSORcnt) ----------------
#if __has_builtin(__builtin_amdgcn_tensor_load_to_lds_d2) || __has_builtin(__builtin_amdgcn_tensor_load_to_lds)
#define HAVE_TDM 1
#else
#define HAVE_TDM 0
#endif

typedef unsigned int u32x4 __attribute__((ext_vector_type(4)));
typedef int          i32x4 __attribute__((ext_vector_type(4)));
typedef int          i32x8 __attribute__((ext_vector_type(8)));

#if HAVE_TDM
// Load a [tile_rows x 256] f32 tile (row stride 256 elems) from g_src into LDS.
// Rows >= tensor_rows read as zero (TDM OOB rule) -- used instead of clamping.
__device__ __forceinline__ void tdm_load_tile_2d(float* lds_dst, const float* g_src,
                                                 int tile_rows, int tensor_rows) {
    unsigned           lds_addr = (unsigned)(uintptr_t)lds_dst;            // LDS byte offset
    unsigned long long ga       = (unsigned long long)(uintptr_t)g_src;   // global byte addr

    u32x4 g0;
    g0[0] = 1u;                                               // count=1 (valid user D#)
    g0[1] = lds_addr;                                         // lds_addr [63:32]
    g0[2] = (unsigned)(ga & 0xffffffffu);                     // global_addr lo
    g0[3] = (unsigned)((ga >> 32) & 0x01ffffffu) | (2u << 30);// global_addr[56:32] | type=2

    i32x8 g1;
    g1[0] = (int)(2u << 16);                                  // data_size=2 -> 4 bytes
    g1[1] = (int)(256u << 16);                                // tensor_dim0[15:0]=256
    g1[2] = (int)(((unsigned)tensor_rows & 0xffffu) << 16);   // dim0 hi=0 | tensor_dim1 lo
    g1[3] = (int)((((unsigned)tensor_rows >> 16) & 0xffffu)   // tensor_dim1 hi
                  | (256u << 16));                            // tile_dim0=256
    g1[4] = (int)((unsigned)tile_rows & 0xffffu);             // tile_dim1 | tile_dim2=0
    g1[5] = 256;                                              // tensor_dim0_stride lo
    g1[6] = 0;                                                // stride0 hi | stride1 lo
    g1[7] = 0;                                                // stride1 hi

#if __has_builtin(__builtin_amdgcn_tensor_load_to_lds_d2)
    __builtin_amdgcn_tensor_load_to_lds_d2(g0, g1, 0);
#else
    i32x4 z4 = {0, 0, 0, 0};                                  // groups 2/3 unused (2D tile)
    i32x8 z8 = {0, 0, 0, 0, 0, 0, 0, 0};
    __builtin_amdgcn_tensor_load_to_lds(g0, g1, z4, z4, z8, 0);
#endif
}
#endif

__device__ __forceinline__ void wait_tensor0() {
#if __has_builtin(__builtin_amdgcn_s_wait_tensorcnt)
    __builtin_amdgcn_s_wait_tensorcnt(0);
#else
    asm volatile("s_wait_tensorcnt 0" ::: "memory");
#endif
}

// ---------------- device-wide software barrier (persistent kernel) ----------------
__device__ __forceinline__ void grid_barrier(unsigned* cnt, unsigned* gen, unsigned nb) {
    __builtin_amdgcn_fence(__ATOMIC_RELEASE, "agent");  // publish my stores
    __syncthreads();
    if (threadIdx.x == 0) {
        unsigned g = __hip_atomic_load(gen, __ATOMIC_RELAXED, __HIP_MEMORY_SCOPE_AGENT);
        unsigned a = __hip_atomic_fetch_add(cnt, 1u, __ATOMIC_ACQ_REL, __HIP_MEMORY_SCOPE_AGENT);
        if (a == nb - 1u) {
            __hip_atomic_store(cnt, 0u, __ATOMIC_RELAXED, __HIP_MEMORY_SCOPE_AGENT);
            __hip_atomic_fetch_add(gen, 1u, __ATOMIC_RELEASE, __HIP_MEMORY_SCOPE_AGENT);
        } else {
            while (__hip_atomic_load(gen, __ATOMIC_ACQUIRE, __HIP_MEMORY_SCOPE_AGENT) == g) {
                __builtin_amdgcn_s_sleep(2);
            }
        }
    }
    __syncthreads();
    __builtin_amdgcn_fence(__ATOMIC_ACQUIRE, "agent");  // see others' stores
}

// ---------------- init: seed neighbor-visible planes + barrier state ----------------
// planes layout (8 planes * NELEM floats): ub, vb1, vb2, p1, p2, q11, q22, q12
__global__ void tgv_init(const float* __restrict__ f, float* __restrict__ planes,
                         unsigned* __restrict__ bar) {
    int i = blockIdx.x * blockDim.x + threadIdx.x;
    if (i < NELEM) {
        planes[0 * NELEM + i] = f[i];   // u_bar = u0
        planes[1 * NELEM + i] = 0.f;    // v_bar1
        planes[2 * NELEM + i] = 0.f;    // v_bar2
        planes[3 * NELEM + i] = 0.f;    // p1
        planes[4 * NELEM + i] = 0.f;    // p2
        planes[5 * NELEM + i] = 0.f;    // q11
        planes[6 * NELEM + i] = 0.f;    // q22
        planes[7 * NELEM + i] = 0.f;    // q12
    }
    if (i < 2) bar[i] = 0u;
}

// ---------------- persistent PDHG kernel ----------------
__global__ __launch_bounds__(TPB, 1) void tgv_pdhg(
    const float* __restrict__ f, const float* __restrict__ rp,
    const int* __restrict__ Tp, float* __restrict__ out,
    float* planes, unsigned* bar)
{
    __shared__ float lds[5 * STG];  // 25.6 KB of the 320 KB/WGP

    const int tid = threadIdx.x;
    const int blk = blockIdx.x;
    const int img = blk >> 6;               // 64 strips per image
    const int r0  = (blk & 63) * ROWS;      // first row of this strip
    const int base = img * HW + r0 * Ww;
    const int x = tid;                      // column == lane-friendly coalescing

    float* ub  = planes + 0 * NELEM;
    float* vb1 = planes + 1 * NELEM;
    float* vb2 = planes + 2 * NELEM;
    float* p1  = planes + 3 * NELEM;
    float* p2  = planes + 4 * NELEM;
    float* q11 = planes + 5 * NELEM;
    float* q22 = planes + 6 * NELEM;
    float* q12 = planes + 7 * NELEM;

    const float alpha0 = rp[0];
    const float alpha1 = rp[1];
    const int   T      = Tp[0];
    const float tau    = 0.28867513459481287f;  // 1/sqrt(12)
    const float sigma  = tau;
    const float i1pt   = 1.0f / (1.0f + tau);
    const float ia0    = 1.0f / alpha0;
    const float ia1    = 1.0f / alpha1;

    // register-resident per-thread state (never visible to other threads: u, v, f;
    // p,q old values are register-carried, halos come from L2 via LDS staging)
    float u_r[ROWS], v1_r[ROWS], v2_r[ROWS], f_r[ROWS];
    float pr1[ROWS], pr2[ROWS], qr11[ROWS], qr22[ROWS], qr12[ROWS];
#pragma unroll
    for (int k = 0; k < ROWS; ++k) {
        int gi = base + k * Ww + x;
        f_r[k] = f[gi];
        u_r[k] = f_r[k];
        v1_r[k] = 0.f; v2_r[k] = 0.f;
        pr1[k] = 0.f; pr2[k] = 0.f;
        qr11[k] = 0.f; qr22[k] = 0.f; qr12[k] = 0.f;
    }

    for (int t = 0; t < T; ++t) {
        // ===== DUAL: stage u_bar, v_bar (rows r0 .. r0+4) via per-lane async =====
#pragma unroll
        for (int j = 0; j < 5; ++j) {
            int w   = tid + j * TPB;            // 0..1279
            int row = w >> 8, col = w & 255;
            int gy  = r0 + row; gy = gy > (Hh - 1) ? (Hh - 1) : gy;  // clamp (masked anyway)
            int gi  = img * HW + gy * Ww + col;
            async_copy_f32(&lds[0 * STG + w], &ub[gi]);
            async_copy_f32(&lds[1 * STG + w], &vb1[gi]);
            async_copy_f32(&lds[2 * STG + w], &vb2[gi]);
        }
        wait_async0();
        __syncthreads();

#pragma unroll
        for (int k = 0; k < ROWS; ++k) {
            const int y  = r0 + k;
            const int li = k * Ww + x;
            const int gi = base + li;
            const bool ym = (y < Hh - 1);
            const bool xm = (x < Ww - 1);

            float ubc  = lds[0 * STG + li];
            float vb1c = lds[1 * STG + li];
            float vb2c = lds[2 * STG + li];

            // p <- proj_{alpha1}( p + sigma*(grad(u_bar) - v_bar) )
            float g1 = ym ? (lds[0 * STG + li + Ww] - ubc) : 0.f;
            float g2 = xm ? (lds[0 * STG + li + 1]  - ubc) : 0.f;
            float pt1 = pr1[k] + sigma * (g1 - vb1c);
            float pt2 = pr2[k] + sigma * (g2 - vb2c);
            float n = sqrtf(pt1 * pt1 + pt2 * pt2);
            float s = fmaxf(1.f, n * ia1);
            pt1 /= s; pt2 /= s;
            pr1[k] = pt1; pr2[k] = pt2;
            p1[gi] = pt1; p2[gi] = pt2;

            // q <- proj_{alpha0}( q + sigma*sym_grad(v_bar) ), weight 2 on e12
            float e11 = ym ? (lds[1 * STG + li + Ww] - vb1c) : 0.f;
            float e22 = xm ? (lds[2 * STG + li + 1]  - vb2c) : 0.f;
            float a12 = xm ? (lds[1 * STG + li + 1]  - vb1c) : 0.f;
            float b12 = ym ? (lds[2 * STG + li + Ww] - vb2c) : 0.f;
            float e12 = 0.5f * (a12 + b12);
            float t11 = qr11[k] + sigma * e11;
            float t22 = qr22[k] + sigma * e22;
            float t12 = qr12[k] + sigma * e12;
            n = sqrtf(t11 * t11 + t22 * t22 + 2.f * t12 * t12);
            s = fmaxf(1.f, n * ia0);
            t11 /= s; t22 /= s; t12 /= s;
            qr11[k] = t11; qr22[k] = t22; qr12[k] = t12;
            q11[gi] = t11; q22[gi] = t22; q12[gi] = t12;
        }

        grid_barrier(&bar[0], &bar[1], gridDim.x);   // new p,q visible device-wide

        // ===== PRIMAL: stage p,q (rows r0-1 .. r0+3) via TDM 2D tile loads =====
#if HAVE_TDM
        if (tid == 0) {
            // For r0==0 shift LDS down one row; the "row -1" slot is never read (y>0 mask).
            const int gy0  = (r0 == 0) ? 0 : r0 - 1;
            const int lrow = (r0 == 0) ? 1 : 0;
            const int trows = 5 - lrow;
            const int tens_rows = Hh - gy0;     // OOB rows (none here) would read 0
            const int gb = img * HW + gy0 * Ww;
            tdm_load_tile_2d(&lds[0 * STG + lrow * Ww], &p1[gb],  trows, tens_rows);
            tdm_load_tile_2d(&lds[1 * STG + lrow * Ww], &p2[gb],  trows, tens_rows);
            tdm_load_tile_2d(&lds[2 * STG + lrow * Ww], &q11[gb], trows, tens_rows);
            tdm_load_tile_2d(&lds[3 * STG + lrow * Ww], &q22[gb], trows, tens_rows);
            tdm_load_tile_2d(&lds[4 * STG + lrow * Ww], &q12[gb], trows, tens_rows);
            wait_tensor0();
        }
        __syncthreads();
#else
#pragma unroll
        for (int j = 0; j < 5; ++j) {
            int w   = tid + j * TPB;
            int row = w >> 8, col = w & 255;
            int gy  = r0 + row - 1; gy = gy < 0 ? 0 : gy;  // clamp (masked anyway)
            int gi  = img * HW + gy * Ww + col;
            async_copy_f32(&lds[0 * STG + w], &p1[gi]);
            async_copy_f32(&lds[1 * STG + w], &p2[gi]);
            async_copy_f32(&lds[2 * STG + w], &q11[gi]);
            async_copy_f32(&lds[3 * STG + w], &q22[gi]);
            async_copy_f32(&lds[4 * STG + w], &q12[gi]);
        }
        wait_async0();
        __syncthreads();
#endif

#pragma unroll
        for (int k = 0; k < ROWS; ++k) {
            const int y  = r0 + k;
            const int lc = (k + 1) * Ww + x;   // center (staged rows start at r0-1)
            const int lu = k * Ww + x;         // y-1
            const bool ymh = (y < Hh - 1), yml = (y > 0);
            const bool xmh = (x < Ww - 1), xml = (x > 0);

            // u <- (u + tau*div(p) + tau*f) / (1+tau)
            float d1 = (ymh ? pr1[k] : 0.f) - (yml ? lds[0 * STG + lu] : 0.f);
            float d2 = (xmh ? pr2[k] : 0.f) - (xml ? lds[1 * STG + lc - 1] : 0.f);
            float un = (u_r[k] + tau * (d1 + d2) + tau * f_r[k]) * i1pt;

            // v <- v + tau*(p + sym_div(q))
            float c1 = (ymh ? qr11[k] : 0.f) - (yml ? lds[2 * STG + lu] : 0.f)
                     + (xmh ? qr12[k] : 0.f) - (xml ? lds[4 * STG + lc - 1] : 0.f);
            float c2 = (ymh ? qr12[k] : 0.f) - (yml ? lds[4 * STG + lu] : 0.f)
                     + (xmh ? qr22[k] : 0.f) - (xml ? lds[3 * STG + lc - 1] : 0.f);
            float v1n = v1_r[k] + tau * (pr1[k] + c1);
            float v2n = v2_r[k] + tau * (pr2[k] + c2);

            // over-relaxation, publish bars for next iteration's dual step
            const int gi = base + k * Ww + x;
            ub[gi]  = 2.f * un  - u_r[k];
            vb1[gi] = 2.f * v1n - v1_r[k];
            vb2[gi] = 2.f * v2n - v2_r[k];
            u_r[k] = un; v1_r[k] = v1n; v2_r[k] = v2n;
        }

        grid_barrier(&bar[0], &bar[1], gridDim.x);   // bars visible device-wide
    }

#pragma unroll
    for (int k = 0; k < ROWS; ++k) out[base + k * Ww + x] = u_r[k];
}

// ---------------- host entry ----------------
extern "C" void kernel_launch(void* const* d_in, const int* in_sizes, int n_in,
                              void* d_out, int out_size, void* d_ws, size_t ws_size,
                              hipStream_t stream) {
    (void)in_sizes; (void)n_in; (void)out_size; (void)ws_size;
    const float* f  = (const float*)d_in[0];  // u0, (2,256,256) f32
    const float* rp = (const float*)d_in[1];  // [alpha0, alpha1]
    const int*   Tp = (const int*)d_in[2];    // T = 128
    float* out = (float*)d_out;

    // ws layout: [0,256) barrier counters; then 8 planes * NELEM f32 (~4.2 MB)
    unsigned* bar = (unsigned*)d_ws;
    float* planes = (float*)((char*)d_ws + 256);

    tgv_init<<<NELEM / TPB, TPB, 0, stream>>>(f, planes, bar);
    tgv_pdhg<<<NBLK, TPB, 0, stream>>>(f, rp, Tp, out, planes, bar);
}